// CrossModalTripletLoss_error_16011638080289
// MI455X (gfx1250) — compile-verified
//
#include <hip/hip_runtime.h>
#include <hip/hip_bf16.h>
#include <math.h>

// Problem constants (match reference)
#define B_N 4096   // batch
#define D_K 128    // feature dim
#define C_N 80     // classes
// MARGIN = 1.0, ERROR_NUM = 4

typedef __attribute__((ext_vector_type(2))) float v2f;
typedef __attribute__((ext_vector_type(8))) float v8f;
typedef __attribute__((ext_vector_type(4))) unsigned v4u;
typedef __attribute__((ext_vector_type(8))) int v8i;
typedef __attribute__((ext_vector_type(4))) int v4i;

// ---------------------------------------------------------------------------
// Threefry-2x32 (20 rounds) — matches jax._src.prng.threefry2x32
// ---------------------------------------------------------------------------
__host__ __device__ __forceinline__ unsigned tf_rotl(unsigned v, int r) {
  return (v << r) | (v >> (32 - r));
}

__host__ __device__ __forceinline__ void threefry2x32(unsigned k0, unsigned k1,
                                                      unsigned x0, unsigned x1,
                                                      unsigned& o0, unsigned& o1) {
  const unsigned ks2 = k0 ^ k1 ^ 0x1BD11BDAu;
  const unsigned ks[3] = {k0, k1, ks2};
  const int R[2][4] = {{13, 15, 26, 6}, {17, 29, 16, 24}};
  x0 += ks[0];
  x1 += ks[1];
#pragma unroll
  for (int g = 0; g < 5; ++g) {
#pragma unroll
    for (int r = 0; r < 4; ++r) {
      x0 += x1;
      x1 = tf_rotl(x1, R[g & 1][r]);
      x1 ^= x0;
    }
    x0 += ks[(g + 1) % 3];
    x1 += ks[(g + 2) % 3] + (unsigned)(g + 1);
  }
  o0 = x0;
  o1 = x1;
}

// jax uniform [0,1): (bits >> 9) | 0x3f800000 bitcast f32, minus 1
__device__ __forceinline__ float tf_uniform(unsigned bits) {
  return __uint_as_float((bits >> 9) | 0x3f800000u) - 1.0f;
}

// ---------------------------------------------------------------------------
// Kernel 1: row squared-norms + one-hot -> label id
// ---------------------------------------------------------------------------
__global__ void cmtl_prep_kernel(const float* __restrict__ img,
                                 const float* __restrict__ txt,
                                 const float* __restrict__ labels,
                                 float* __restrict__ r_img,
                                 float* __restrict__ r_txt,
                                 int* __restrict__ lab) {
  int i = blockIdx.x * blockDim.x + threadIdx.x;
  if (i >= B_N) return;
  const float4* pi = (const float4*)(img + (size_t)i * D_K);
  const float4* pt = (const float4*)(txt + (size_t)i * D_K);
  float si = 0.f, st = 0.f;
#pragma unroll 8
  for (int k = 0; k < D_K / 4; ++k) {
    float4 a = pi[k];
    float4 b = pt[k];
    si += a.x * a.x + a.y * a.y + a.z * a.z + a.w * a.w;
    st += b.x * b.x + b.y * b.y + b.z * b.z + b.w * b.w;
  }
  r_img[i] = si;
  r_txt[i] = st;
  int l = 0;
  for (int c = 0; c < C_N; ++c)
    if (labels[(size_t)i * C_N + c] > 0.5f) l = c;
  lab[i] = l;
}

// ---------------------------------------------------------------------------
// Kernel 2: per-row top-4 of threefry uniforms over valid negatives.
// One wave (32 lanes) per row; lane-local top-4 then shfl_xor merge.
// Tie rule matches lax.top_k: value desc, then lower index first.
// ---------------------------------------------------------------------------
__global__ void cmtl_sample_kernel(const int* __restrict__ lab,
                                   unsigned k0, unsigned k1,
                                   int* __restrict__ out_idx) {
  int w = (int)((blockIdx.x * (unsigned)blockDim.x + threadIdx.x) >> 5);  // row
  int lane = (int)(threadIdx.x & 31u);
  int labi = lab[w];

  float v[4] = {-INFINITY, -INFINITY, -INFINITY, -INFINITY};
  int id[4] = {0x7FFFFFFF, 0x7FFFFFFF, 0x7FFFFFFF, 0x7FFFFFFF};

  unsigned base = (unsigned)w * 4096u;
  for (int j = lane; j < B_N; j += 32) {
    unsigned y0, y1;
    threefry2x32(k0, k1, 0u, base + (unsigned)j, y0, y1);
    float u = tf_uniform(y0 ^ y1);  // partitionable 32-bit path: x0^x1
    float s = (lab[j] != labi) ? u : -INFINITY;
    if (s > v[3] || (s == v[3] && j < id[3])) {
      v[3] = s;
      id[3] = j;
#pragma unroll
      for (int k = 3; k > 0; --k) {
        bool sw = (v[k] > v[k - 1]) || (v[k] == v[k - 1] && id[k] < id[k - 1]);
        if (sw) {
          float tv = v[k]; v[k] = v[k - 1]; v[k - 1] = tv;
          int ti = id[k]; id[k] = id[k - 1]; id[k - 1] = ti;
        }
      }
    }
  }

  // wave32 merge of sorted-4 lists
#pragma unroll
  for (int off = 16; off > 0; off >>= 1) {
    float ov[4]; int oid[4];
#pragma unroll
    for (int k = 0; k < 4; ++k) {
      ov[k] = __shfl_xor(v[k], off, 32);
      oid[k] = __shfl_xor(id[k], off, 32);
    }
    float mv[4]; int mid[4];
    int p = 0, q = 0;
#pragma unroll
    for (int k = 0; k < 4; ++k) {
      bool takeA = (v[p] > ov[q]) || (v[p] == ov[q] && id[p] < oid[q]);
      if (takeA) { mv[k] = v[p]; mid[k] = id[p]; ++p; }
      else       { mv[k] = ov[q]; mid[k] = oid[q]; ++q; }
    }
#pragma unroll
    for (int k = 0; k < 4; ++k) { v[k] = mv[k]; id[k] = mid[k]; }
  }

  if (lane == 0) {
#pragma unroll
    for (int k = 0; k < 4; ++k) out_idx[w * 4 + k] = id[k];
  }
}

// ---------------------------------------------------------------------------
// Kernel 3: G = image @ text.T in full fp32 via V_WMMA_F32_16X16X4_F32.
//
// Block = 8 waves. Block owns a 64-row M super-tile; its 64x128 A panel
// (32 KB) is staged once into LDS by the Tensor Data Mover (one wave issues
// tensor_load_to_lds, waits TENSORcnt==0, then a workgroup barrier releases
// the readers). Each wave computes a 64(M)x32(N) tile: 4 A-frags (ds_load_b64)
// x 2 B-frags (global_load_b64) -> 8 chained f32 WMMAs per K-step of 4.
// acc = 4x2 v8f = 64 VGPRs; __launch_bounds__(256,1) gives the allocator a
// full VGPR budget so accumulators stay in registers (round-3 disasm showed
// scratch spills around every wmma with the 4x4 tile).
//
// A 16x4 frag layout: lane m=l&15 holds (K=2*kh, K=2*kh+1), kh = l>>4.
// C/D 16x16 layout: VGPR v holds row M = v + 8*kh, col N = l&15.
// ---------------------------------------------------------------------------
__global__ void __launch_bounds__(256, 1)
cmtl_gemm_kernel(const float* __restrict__ img,
                 const float* __restrict__ txt,
                 float* __restrict__ G) {
  __shared__ float smA[64 * D_K];  // 32 KB A panel for this block

  const int bm = (int)blockIdx.x >> 4;   // 64 M super-tiles
  const int bn = (int)blockIdx.x & 15;   // 16 N super-tiles (of 256 cols)
  const int wave = (int)(threadIdx.x >> 5);
  const int lane = (int)(threadIdx.x & 31u);
  const int r = lane & 15;
  const int kh = lane >> 4;

  const int tmb = bm * 64;               // M base of the block's A panel
  const int tnb = (bn * 8 + wave) * 32;  // N base of this wave's tile

  // ---- Stage the 64x128 fp32 A panel into LDS via TDM ------------------
#if __has_builtin(__builtin_amdgcn_tensor_load_to_lds)
  if (threadIdx.x < 32) {
    unsigned lds_addr = (unsigned)(size_t)(void*)&smA[0];
    unsigned long long ga =
        (unsigned long long)(const void*)(img + (size_t)tmb * D_K);
    // D# group0: [1:0]=count=1, [31]=gather off; lds_addr; 57b global_addr;
    //            [127:126]=type=2 ("image")
    v4u g0 = {1u, lds_addr, (unsigned)(ga & 0xFFFFFFFFu),
              (unsigned)((ga >> 32) & 0x01FFFFFFu) | 0x80000000u};
    // D# group1: data_size=2 (4B); tensor_dim0=128 @bit48; tensor_dim1=64
    //            @bit80; tile_dim0=128 @bit112; tile_dim1=64 @bit128;
    //            tensor_dim0_stride=128 @bit160; everything else 0.
    v8i g1 = {0x00020000, 0x00800000, 0x00400000, 0x00800000, 64, 128, 0, 0};
    v4i g2 = {0, 0, 0, 0};  // 2-D tile: groups 2/3 unused
    v4i g3 = {0, 0, 0, 0};
    v8i g4 = {0, 0, 0, 0, 0, 0, 0, 0};  // clang-23 6-arg form: extra group
    __builtin_amdgcn_tensor_load_to_lds(g0, g1, g2, g3, g4, 0);
    __builtin_amdgcn_s_wait_tensorcnt(0);
  }
#else
  for (int t = (int)threadIdx.x; t < 64 * D_K / 4; t += 256)
    ((float4*)smA)[t] = ((const float4*)(img + (size_t)tmb * D_K))[t];
#endif
  __syncthreads();

  // ---- 64x32 per-wave WMMA tile ----------------------------------------
  const float* aBase = &smA[r * D_K + 2 * kh];
  const float* brow[2];
#pragma unroll
  for (int ni = 0; ni < 2; ++ni) {
    brow[ni] = txt + (size_t)(tnb + ni * 16 + r) * D_K + 2 * kh;
    __builtin_prefetch(brow[ni], 0, 3);
  }

  v8f acc[4][2];
#pragma unroll
  for (int mi = 0; mi < 4; ++mi)
#pragma unroll
    for (int ni = 0; ni < 2; ++ni)
      acc[mi][ni] = (v8f){0.f, 0.f, 0.f, 0.f, 0.f, 0.f, 0.f, 0.f};

#pragma unroll 4
  for (int kb = 0; kb < D_K; kb += 4) {
    v2f a[4], b[2];
#pragma unroll
    for (int mi = 0; mi < 4; ++mi)
      a[mi] = *(const v2f*)(aBase + mi * 16 * D_K + kb);  // ds_load_b64
#pragma unroll
    for (int ni = 0; ni < 2; ++ni)
      b[ni] = *(const v2f*)(brow[ni] + kb);               // global_load_b64
#pragma unroll
    for (int mi = 0; mi < 4; ++mi)
#pragma unroll
      for (int ni = 0; ni < 2; ++ni)
        // 8 args: (neg_a, A, neg_b, B, c_mod, C, reuse_a, reuse_b)
        acc[mi][ni] = __builtin_amdgcn_wmma_f32_16x16x4_f32(
            false, a[mi], false, b[ni], (short)0, acc[mi][ni], false, false);
  }

#pragma unroll
  for (int mi = 0; mi < 4; ++mi) {
#pragma unroll
    for (int ni = 0; ni < 2; ++ni) {
      float* gout = G + (size_t)(tmb + mi * 16 + 8 * kh) * B_N +
                    (size_t)(tnb + ni * 16 + r);
#pragma unroll
      for (int vv = 0; vv < 8; ++vv) gout[(size_t)vv * B_N] = acc[mi][ni][vv];
    }
  }
}

// ---------------------------------------------------------------------------
// Kernel 4: gather 9 distances per row, relu-margin, deterministic reduce.
// loss = mean(relu(pos - neg_it + 1)) + mean(relu(pos - neg_ti + 1))
// dist_ti[i,j] = sqrt(r_txt[i] + r_img[j] - 2*G[j,i])   (G = img @ txt.T)
// ---------------------------------------------------------------------------
__global__ void cmtl_loss_kernel(const float* __restrict__ G,
                                 const float* __restrict__ r_img,
                                 const float* __restrict__ r_txt,
                                 const int* __restrict__ idx_it,
                                 const int* __restrict__ idx_ti,
                                 float* __restrict__ out) {
  __shared__ float sm[256];
  int tid = threadIdx.x;
  float acc = 0.f;
  for (int i = tid; i < B_N; i += 256) {
    float ri = r_img[i];
    float ti = r_txt[i];
    float pos = sqrtf(fmaxf(ri + ti - 2.f * G[(size_t)i * B_N + i], 0.f));
#pragma unroll
    for (int k = 0; k < 4; ++k) {
      int j = idx_it[i * 4 + k];
      float d = sqrtf(fmaxf(ri + r_txt[j] - 2.f * G[(size_t)i * B_N + j], 0.f));
      acc += fmaxf(pos - d + 1.0f, 0.f);
    }
#pragma unroll
    for (int k = 0; k < 4; ++k) {
      int j = idx_ti[i * 4 + k];
      float d = sqrtf(fmaxf(ti + r_img[j] - 2.f * G[(size_t)j * B_N + i], 0.f));
      acc += fmaxf(pos - d + 1.0f, 0.f);
    }
  }
  sm[tid] = acc;
  __syncthreads();
  for (int s = 128; s > 0; s >>= 1) {
    if (tid < s) sm[tid] += sm[tid + s];
    __syncthreads();
  }
  if (tid == 0) out[0] = sm[0] * (1.0f / 16384.0f);  // /(B*ERROR_NUM) per term
}

// ---------------------------------------------------------------------------
extern "C" void kernel_launch(void* const* d_in, const int* in_sizes, int n_in,
                              void* d_out, int out_size, void* d_ws,
                              size_t ws_size, hipStream_t stream) {
  const float* img = (const float*)d_in[0];     // [4096,128]
  const float* txt = (const float*)d_in[1];     // [4096,128]
  const float* labels = (const float*)d_in[2];  // [4096,80] one-hot
  float* out = (float*)d_out;

  // Workspace layout (needs ~66 MB; Gram matrix is L2-resident: 64MB < 192MB)
  char* ws = (char*)d_ws;
  float* G = (float*)ws;
  size_t off = (size_t)B_N * B_N * sizeof(float);
  float* r_img = (float*)(ws + off); off += B_N * sizeof(float);
  float* r_txt = (float*)(ws + off); off += B_N * sizeof(float);
  int* lab = (int*)(ws + off);       off += B_N * sizeof(int);
  int* idx_it = (int*)(ws + off);    off += (size_t)B_N * 4 * sizeof(int);
  int* idx_ti = (int*)(ws + off);

  // Host-side key derivation: key(42) = (0,42); fold-like split (partitionable):
  // ks1 = threefry(key, (0,0)), ks2 = threefry(key, (0,1))
  unsigned s1k0, s1k1, s2k0, s2k1;
  threefry2x32(0u, 42u, 0u, 0u, s1k0, s1k1);
  threefry2x32(0u, 42u, 0u, 1u, s2k0, s2k1);

  cmtl_prep_kernel<<<B_N / 256, 256, 0, stream>>>(img, txt, labels, r_img,
                                                  r_txt, lab);
  // one wave per row -> 4096 waves -> 512 blocks of 256 threads
  cmtl_sample_kernel<<<(B_N * 32) / 256, 256, 0, stream>>>(lab, s1k0, s1k1,
                                                           idx_it);
  cmtl_sample_kernel<<<(B_N * 32) / 256, 256, 0, stream>>>(lab, s2k0, s2k1,
                                                           idx_ti);
  // 64 M-supertiles x 16 N-supertiles; 8 waves/block, one 64x32 tile each
  cmtl_gemm_kernel<<<1024, 256, 0, stream>>>(img, txt, G);
  cmtl_loss_kernel<<<1, 256, 0, stream>>>(G, r_img, r_txt, idx_it, idx_ti, out);
}